// GGC2LW_10599979286673
// MI455X (gfx1250) — compile-verified
//
#include <hip/hip_runtime.h>
#include <hip/hip_bf16.h>
#include <cmath>

#define B_  64
#define N_  512
#define P_  256
#define E_  4096
#define F_  32
#define NT_ 8
#define H_  64
#define W_  8
#define G_  64
#define RH_ 256
#define H1_ 128
#define Q_  9

typedef float v2f __attribute__((ext_vector_type(2)));
typedef float v8f __attribute__((ext_vector_type(8)));
typedef unsigned int u32x4 __attribute__((ext_vector_type(4)));
typedef int i32x4 __attribute__((ext_vector_type(4)));
typedef int i32x8 __attribute__((ext_vector_type(8)));

// D = A(16x4) * B(4x16) + C, exact fp32 path (V_WMMA_F32_16X16X4_F32).
__device__ __forceinline__ v8f wmma4(v2f a, v2f b, v8f c) {
  return __builtin_amdgcn_wmma_f32_16x16x4_f32(false, a, false, b, (short)0, c,
                                               false, false);
}

// ---------------------------------------------------------------- indices
__global__ void k_indices(const float* __restrict__ linc,
                          const float* __restrict__ rinc,
                          int* __restrict__ lidx, int* __restrict__ ridx) {
  int e = blockIdx.x * blockDim.x + threadIdx.x;
  if (e >= E_) return;
  int li = 0;
  for (int p = 0; p < P_; ++p) li += (linc[(size_t)p * E_ + e] > 0.5f) ? p : 0;
  lidx[e] = li;
  int ri = 0;
  for (int n = 0; n < N_; ++n) ri += (rinc[(size_t)e * N_ + n] > 0.5f) ? n : 0;
  ridx[e] = ri;
}

// ---------------------------------------------------------------- CSR build
// Thread p scans e in ascending order -> deterministic bucket order.
#define ECAP 96
__global__ __launch_bounds__(256) void k_csr(const int* __restrict__ lidx,
                                             int* __restrict__ ecnt,
                                             int* __restrict__ elist) {
  int p = threadIdx.x;
  int cnt = 0;
  for (int e = 0; e < E_; ++e) {
    if (lidx[e] == p) {
      if (cnt < ECAP) elist[p * ECAP + cnt] = e;
      ++cnt;
    }
  }
  ecnt[p] = (cnt > ECAP) ? ECAP : cnt;
}

// ---------------------------------------------------------------- grouped conv
// One workgroup per group n: X1g(64x128) @ Wgᵀ(128x32), bias+relu fused.
// LDS tiles staged by the Tensor Data Mover with D# padding (2 dwords / 128)
// so WMMA fragment reads are LDS-bank-conflict-free (row stride 130).
#define SAP 130
__global__ __launch_bounds__(128) void k_conv(const float* __restrict__ x1,
                                              const float* __restrict__ conv_w,
                                              const float* __restrict__ conv_b,
                                              float* __restrict__ x) {
  __shared__ float sA[64 * SAP];  // [batch][k] padded
  __shared__ float sB[32 * SAP];  // [f][k] padded
  const int n = blockIdx.x;
  const int tid = threadIdx.x;

#if __has_builtin(__builtin_amdgcn_tensor_load_to_lds) && \
    __has_builtin(__builtin_amdgcn_s_wait_tensorcnt)
  if (tid < 32) {  // one wave issues the two TDM descriptors
    i32x4 z4 = {0, 0, 0, 0};
    i32x8 z8 = {0, 0, 0, 0, 0, 0, 0, 0};
    // A tile: 2D, tile 128x64, row stride 65536 elems, 4B data, pad 2dw/128dw
    {
      unsigned long long ga =
          (unsigned long long)(const void*)(x1 + (size_t)n * 128);
      unsigned int lds = (unsigned int)(unsigned long long)(void*)&sA[0];
      u32x4 g0 = {1u, lds, (unsigned)(ga & 0xFFFFFFFFu),
                  (unsigned)(((ga >> 32) & 0x1FFFFFFu) | 0x80000000u)};
      i32x8 g1 = {(int)0x3920000,   // data_size=4B, pad_en, amt=2dw, intv=128dw
                  (int)0x800000,    // tensor_dim0 = 128 (bits 79:48)
                  (int)0x400000,    // tensor_dim1 = 64
                  (int)0x800000,    // tile_dim0 = 128
                  64,               // tile_dim1 = 64
                  65536, 0, 0};     // tensor_dim0_stride = 65536
      __builtin_amdgcn_tensor_load_to_lds(g0, g1, z4, z4, z8, 0);
    }
    // B tile: 1D, 4096 contiguous elems (32 rows x 128), same padding
    {
      unsigned long long ga =
          (unsigned long long)(const void*)(conv_w + (size_t)n * 4096);
      unsigned int lds = (unsigned int)(unsigned long long)(void*)&sB[0];
      u32x4 g0 = {1u, lds, (unsigned)(ga & 0xFFFFFFFFu),
                  (unsigned)(((ga >> 32) & 0x1FFFFFFu) | 0x80000000u)};
      i32x8 g1 = {(int)0x3920000,
                  (int)0x10000000,  // tensor_dim0 = 4096
                  (int)0x10000,     // tensor_dim1 = 1
                  (int)0x10000000,  // tile_dim0 = 4096
                  0, 4096, 0, 0};
      __builtin_amdgcn_tensor_load_to_lds(g0, g1, z4, z4, z8, 0);
    }
    __builtin_amdgcn_s_wait_tensorcnt(0);
  }
#else
  for (int i = tid; i < 64 * 128; i += 128) {
    int m = i >> 7, k = i & 127;
    sA[m * SAP + k] = x1[(size_t)m * 65536 + (size_t)n * 128 + k];
  }
  for (int i = tid; i < 32 * 128; i += 128) {
    int f = i >> 7, k = i & 127;
    sB[f * SAP + k] = conv_w[(size_t)(n * 32 + f) * 128 + k];
  }
#endif
  __syncthreads();

  const int wv = tid >> 5, lane = tid & 31, lm = lane & 15, kh = lane >> 4;
  const int m0 = wv * 16;
  v8f c0 = {}, c1 = {};
#pragma unroll
  for (int kk = 0; kk < 32; ++kk) {
    int kb = kk * 4 + 2 * kh;
    v2f a;  a.x  = sA[(m0 + lm) * SAP + kb]; a.y = sA[(m0 + lm) * SAP + kb + 1];
    v2f b0; b0.x = sB[lm * SAP + kb];        b0.y = sB[lm * SAP + kb + 1];
    v2f b1; b1.x = sB[(lm + 16) * SAP + kb]; b1.y = sB[(lm + 16) * SAP + kb + 1];
    c0 = wmma4(a, b0, c0);
    c1 = wmma4(a, b1, c1);
  }
#pragma unroll
  for (int r = 0; r < 8; ++r) {
    int m = m0 + r + 8 * kh;
    int f0 = lm, f1 = lm + 16;
    float v0 = c0[r] + conv_b[n * 32 + f0];
    float v1 = c1[r] + conv_b[n * 32 + f1];
    x[((size_t)m * N_ + n) * F_ + f0] = fmaxf(v0, 0.f);
    x[((size_t)m * N_ + n) * F_ + f1] = fmaxf(v1, 0.f);
  }
}

// ---------------------------------------------------------------- small precomputes
__global__ void k_pre(const float* __restrict__ ef, const float* __restrict__ k1_w,
                      const float* __restrict__ k1_b, const float* __restrict__ x2,
                      const float* __restrict__ k2_w, const float* __restrict__ cap,
                      const float* __restrict__ r2_w, float* __restrict__ kern0,
                      float* __restrict__ k2x2, float* __restrict__ capr2) {
  int gid = blockIdx.x * blockDim.x + threadIdx.x;
  if (gid < E_ * H_) {
    int e = gid >> 6, hh = gid & 63;
    float acc = k1_b[hh];
#pragma unroll
    for (int d = 0; d < 8; ++d) acc += ef[e * 9 + d] * k1_w[hh * 8 + d];
    kern0[gid] = acc;
  }
  if (gid < B_ * H_) {
    int b = gid >> 6, hh = gid & 63;
    float acc = 0.f;
#pragma unroll
    for (int t = 0; t < 8; ++t) acc += x2[b * 8 + t] * k2_w[hh * 8 + t];
    k2x2[gid] = acc;
  }
  if (gid < RH_) {
    float acc = 0.f;
    for (int p = 0; p < P_; ++p) acc += cap[p] * r2_w[gid * P_ + p];
    capr2[gid] = acc;
  }
}

// ---------------------------------------------------------------- E-stage
__global__ __launch_bounds__(128) void k_estage(
    const float* __restrict__ x, const float* __restrict__ kern0,
    const float* __restrict__ k2x2, const float* __restrict__ k3_w,
    const float* __restrict__ k4_w, const float* __restrict__ k5_w,
    const float* __restrict__ k5_b, const int* __restrict__ lidx,
    const int* __restrict__ ridx, const int* __restrict__ pidx,
    float* __restrict__ kernW) {
  __shared__ float sK3[32 * 64];      // [k][h]
  __shared__ float sK4[32 * 64];      // [k][h]
  __shared__ float sK5[64 * 16];      // [k][n], n>=8 zero-padded
  __shared__ float sXP[4][16 * 32];
  __shared__ float sRX[4][16 * 32];
  __shared__ float sKC[4][16 * 64];

  const int b = blockIdx.x >> 6;
  const int tb = blockIdx.x & 63;
  const int tid = threadIdx.x;
  const int wv = tid >> 5, lane = tid & 31, lm = lane & 15, kh = lane >> 4;
  const int e0 = (tb * 4 + wv) * 16;

  for (int i = tid; i < 2048; i += 128) {
    int hh = i >> 5, k = i & 31;
    sK3[k * 64 + hh] = k3_w[hh * 32 + k];
    sK4[k * 64 + hh] = k4_w[hh * 32 + k];
  }
  for (int i = tid; i < 1024; i += 128) {
    int k = i & 63, nn = i >> 6;
    sK5[k * 16 + nn] = (nn < 8) ? k5_w[nn * 64 + k] : 0.f;
  }
  __syncthreads();

  for (int i = 0; i < 16; ++i) {
    int e = e0 + i;
    int pi = pidx[lidx[e]];
    int re = ridx[e];
    sXP[wv][i * 32 + lane] = x[((size_t)b * N_ + pi) * F_ + lane];
    sRX[wv][i * 32 + lane] = x[((size_t)b * N_ + re) * F_ + lane];
  }
  __syncthreads();

  v8f c[4];
#pragma unroll
  for (int nt = 0; nt < 4; ++nt) {
#pragma unroll
    for (int r = 0; r < 8; ++r) {
      int m = r + 8 * kh, hh = nt * 16 + lm;
      c[nt][r] = kern0[(size_t)(e0 + m) * H_ + hh] + k2x2[b * H_ + hh];
    }
  }
#pragma unroll
  for (int kk = 0; kk < 8; ++kk) {
    int kb = kk * 4 + 2 * kh;
    v2f a; a.x = sXP[wv][lm * 32 + kb]; a.y = sXP[wv][lm * 32 + kb + 1];
#pragma unroll
    for (int nt = 0; nt < 4; ++nt) {
      v2f bb; bb.x = sK3[kb * 64 + nt * 16 + lm];
              bb.y = sK3[(kb + 1) * 64 + nt * 16 + lm];
      c[nt] = wmma4(a, bb, c[nt]);
    }
  }
#pragma unroll
  for (int kk = 0; kk < 8; ++kk) {
    int kb = kk * 4 + 2 * kh;
    v2f a; a.x = sRX[wv][lm * 32 + kb]; a.y = sRX[wv][lm * 32 + kb + 1];
#pragma unroll
    for (int nt = 0; nt < 4; ++nt) {
      v2f bb; bb.x = sK4[kb * 64 + nt * 16 + lm];
              bb.y = sK4[(kb + 1) * 64 + nt * 16 + lm];
      c[nt] = wmma4(a, bb, c[nt]);
    }
  }
#pragma unroll
  for (int nt = 0; nt < 4; ++nt) {
#pragma unroll
    for (int r = 0; r < 8; ++r) {
      int m = r + 8 * kh, hh = nt * 16 + lm;
      float v = c[nt][r];
      sKC[wv][m * 64 + hh] = (v > 0.f) ? v : 0.02f * v;  // leaky 0.02
    }
  }
  __syncthreads();

  v8f d = {};
#pragma unroll
  for (int kk = 0; kk < 16; ++kk) {
    int kb = kk * 4 + 2 * kh;
    v2f a;  a.x = sKC[wv][lm * 64 + kb]; a.y = sKC[wv][lm * 64 + kb + 1];
    v2f bb; bb.x = sK5[kb * 16 + lm];    bb.y = sK5[(kb + 1) * 16 + lm];
    d = wmma4(a, bb, d);
  }
  if (lm < 8) {
#pragma unroll
    for (int r = 0; r < 8; ++r) {
      int m = r + 8 * kh;
      float v = fmaxf(d[r] + k5_b[lm], 0.f);
      kernW[((size_t)b * E_ + e0 + m) * W_ + lm] = v;
    }
  }
}

// ---------------------------------------------------------------- agg + gc
// One workgroup per (b, 16-row p-tile): CSR-driven scatter-accumulate into a
// padded LDS tile (race-free: each thread owns two columns; buckets iterate in
// ascending e -> deterministic), then fused gc WMMA GEMM.
#define AP 258
__global__ __launch_bounds__(128) void k_agg(
    const float* __restrict__ x, const float* __restrict__ kernW,
    const int* __restrict__ ridx, const int* __restrict__ ecnt,
    const int* __restrict__ elist, const float* __restrict__ gc_w,
    const float* __restrict__ gc_b, float* __restrict__ yg) {
  __shared__ float aggT[16 * AP];
  const int b = blockIdx.x >> 4;
  const int pt = blockIdx.x & 15;
  const int t = threadIdx.x;
  for (int i = t; i < 16 * AP; i += 128) aggT[i] = 0.f;
  __syncthreads();

  const int c0 = t, c1 = t + 128;
  const int w0 = c0 >> 5, f0 = c0 & 31, w1 = c1 >> 5, f1 = c1 & 31;
  for (int pl = 0; pl < 16; ++pl) {
    int p = pt * 16 + pl;
    int cnt = ecnt[p];
    float a0 = 0.f, a1 = 0.f;
    for (int i = 0; i < cnt; ++i) {
      int e = elist[p * ECAP + i];
      int re = ridx[e];
      const float* kwp = kernW + ((size_t)b * E_ + e) * W_;
      const float* rxp = x + ((size_t)b * N_ + re) * F_;
      a0 += kwp[w0] * rxp[f0];
      a1 += kwp[w1] * rxp[f1];
    }
    aggT[pl * AP + c0] = a0;
    aggT[pl * AP + c1] = a1;
  }
  __syncthreads();

  const int wv = t >> 5, lane = t & 31, lm = lane & 15, kh = lane >> 4;
  v8f c = {};
#pragma unroll 4
  for (int kk = 0; kk < 64; ++kk) {
    int kb = kk * 4 + 2 * kh;
    v2f a; a.x = aggT[lm * AP + kb]; a.y = aggT[lm * AP + kb + 1];
    int g = wv * 16 + lm;
    float2 w2 = *(const float2*)(gc_w + (size_t)g * 256 + kb);
    v2f bb; bb.x = w2.x; bb.y = w2.y;
    c = wmma4(a, bb, c);
  }
#pragma unroll
  for (int r = 0; r < 8; ++r) {
    int p = pt * 16 + r + 8 * kh, g = wv * 16 + lm;
    yg[((size_t)b * P_ + p) * G_ + g] = fmaxf(c[r] + gc_b[g], 0.f);
  }
}

// ---------------------------------------------------------------- head GEMM (split-K x4)
__global__ __launch_bounds__(128) void k_head(const float* __restrict__ yg,
                                              const float* __restrict__ r0_w,
                                              float* __restrict__ hpart) {
  const int seg = blockIdx.x >> 4;
  const int n0 = (blockIdx.x & 15) * 16;
  const int tid = threadIdx.x;
  const int wv = tid >> 5, lane = tid & 31, lm = lane & 15, kh = lane >> 4;
  const int k0 = seg * 4096;
  v8f c = {};
#pragma unroll 4
  for (int kk = 0; kk < 1024; ++kk) {
    int kb = k0 + kk * 4 + 2 * kh;
    float2 a2 = *(const float2*)(yg + (size_t)(16 * wv + lm) * 16384 + kb);
    float2 b2 = *(const float2*)(r0_w + (size_t)(n0 + lm) * 16384 + kb);
    __builtin_prefetch(r0_w + (size_t)(n0 + lm) * 16384 + kb + 256, 0, 1);
    v2f a;  a.x = a2.x;  a.y = a2.y;
    v2f bb; bb.x = b2.x; bb.y = b2.y;
    c = wmma4(a, bb, c);
  }
#pragma unroll
  for (int r = 0; r < 8; ++r) {
    int bb_ = 16 * wv + r + 8 * kh;
    hpart[((size_t)seg * B_ + bb_) * RH_ + n0 + lm] = c[r];
  }
}

__global__ void k_hred(const float* __restrict__ hpart, const float* __restrict__ x2,
                       const float* __restrict__ r1_w, const float* __restrict__ r1_b,
                       const float* __restrict__ capr2, float* __restrict__ h) {
  int idx = blockIdx.x * blockDim.x + threadIdx.x;  // B*RH
  int b = idx >> 8, rh = idx & 255;
  float acc = hpart[(size_t)b * RH_ + rh] +
              hpart[((size_t)B_ + b) * RH_ + rh] +
              hpart[((size_t)2 * B_ + b) * RH_ + rh] +
              hpart[((size_t)3 * B_ + b) * RH_ + rh];
#pragma unroll
  for (int t = 0; t < 8; ++t) acc += x2[b * 8 + t] * r1_w[rh * 8 + t];
  acc += r1_b[rh] + capr2[rh];
  h[idx] = acc;
}

// ---------------------------------------------------------------- wt
__global__ void k_wt(const float* __restrict__ h, const float* __restrict__ r3_w,
                     const float* __restrict__ r3_b, float* __restrict__ wt) {
  int idx = blockIdx.x * blockDim.x + threadIdx.x;  // B*P
  int b = idx >> 8, p = idx & 255;
  float acc = r3_b[p];
  for (int rh = 0; rh < RH_; ++rh) {
    float hv = h[b * RH_ + rh];
    hv = (hv > 0.f) ? hv : 0.01f * hv;  // leaky 0.01
    acc += hv * r3_w[p * RH_ + rh];
  }
  wt[idx] = fmaxf(acc, 0.f);
}

// ---------------------------------------------------------------- final MLP
__global__ __launch_bounds__(128) void k_final(
    const float* __restrict__ wt, const float* __restrict__ yg,
    const float* __restrict__ x2, const float* __restrict__ l1_w,
    const float* __restrict__ l1_b, const float* __restrict__ l3_w,
    const float* __restrict__ l3_b, float* __restrict__ out) {
  __shared__ float zb[72];
  __shared__ float zz[128];
  const int b = blockIdx.x;
  const int t = threadIdx.x;
  if (t < 64) {
    float acc = 0.f;
    for (int p = 0; p < P_; ++p)
      acc += wt[b * P_ + p] * yg[((size_t)b * P_ + p) * G_ + t];
    zb[t] = acc;
  } else if (t < 72) {
    zb[t] = x2[b * NT_ + (t - 64)];
  }
  __syncthreads();
  {
    float acc = l1_b[t];
#pragma unroll
    for (int j = 0; j < 72; ++j) acc += zb[j] * l1_w[t * 72 + j];
    zz[t] = (acc > 0.f) ? acc : (expf(acc) - 1.f);  // elu
  }
  __syncthreads();
  if (t < Q_) {
    float acc = l3_b[t];
    for (int hh = 0; hh < H1_; ++hh) acc += zz[hh] * l3_w[t * H1_ + hh];
    out[b * Q_ + t] = acc;
  }
}

// ---------------------------------------------------------------- launcher
extern "C" void kernel_launch(void* const* d_in, const int* in_sizes, int n_in,
                              void* d_out, int out_size, void* d_ws, size_t ws_size,
                              hipStream_t stream) {
  const float* x1     = (const float*)d_in[0];
  const float* x2     = (const float*)d_in[1];
  const float* conv_w = (const float*)d_in[2];
  const float* conv_b = (const float*)d_in[3];
  const float* k1_w   = (const float*)d_in[4];
  const float* k1_b   = (const float*)d_in[5];
  const float* k2_w   = (const float*)d_in[6];
  const float* k3_w   = (const float*)d_in[7];
  const float* k4_w   = (const float*)d_in[8];
  const float* k5_w   = (const float*)d_in[9];
  const float* k5_b   = (const float*)d_in[10];
  const float* gc_w   = (const float*)d_in[11];
  const float* gc_b   = (const float*)d_in[12];
  const float* r0_w   = (const float*)d_in[13];
  const float* r1_w   = (const float*)d_in[14];
  const float* r1_b   = (const float*)d_in[15];
  const float* r2_w   = (const float*)d_in[16];
  const float* r3_w   = (const float*)d_in[17];
  const float* r3_b   = (const float*)d_in[18];
  const float* l1_w   = (const float*)d_in[19];
  const float* l1_b   = (const float*)d_in[20];
  const float* l3_w   = (const float*)d_in[21];
  const float* l3_b   = (const float*)d_in[22];
  const float* ef     = (const float*)d_in[23];
  const float* linc   = (const float*)d_in[24];
  const float* rinc   = (const float*)d_in[25];
  const float* cap    = (const float*)d_in[26];
  const int*   pidx   = (const int*)d_in[27];
  float* out = (float*)d_out;

  char* ws = (char*)d_ws;
  int*   lidx  = (int*)(ws + 0);
  int*   ridx  = (int*)(ws + 16384);
  float* x     = (float*)(ws + 32768);      // (B,N,F)   4 MiB
  float* kern0 = (float*)(ws + 4227072);    // (E,H)     1 MiB
  float* k2x2  = (float*)(ws + 5275648);    // (B,H)
  float* capr2 = (float*)(ws + 5292032);    // (RH)
  float* kernW = (float*)(ws + 5293056);    // (B,E,W)   8 MiB
  float* yg    = (float*)(ws + 13681664);   // (B,P,G)   4 MiB
  float* h     = (float*)(ws + 17875968);   // (B,RH)
  float* wt    = (float*)(ws + 17941504);   // (B,P)
  int*   ecnt  = (int*)(ws + 18007040);     // (P)
  int*   elist = (int*)(ws + 18008064);     // (P,ECAP)
  float* hpart = (float*)(ws + 18106368);   // (4,B,RH)  256 KiB

  k_indices<<<16, 256, 0, stream>>>(linc, rinc, lidx, ridx);
  k_conv<<<512, 128, 0, stream>>>(x1, conv_w, conv_b, x);
  k_pre<<<1024, 256, 0, stream>>>(ef, k1_w, k1_b, x2, k2_w, cap, r2_w,
                                  kern0, k2x2, capr2);
  k_csr<<<1, 256, 0, stream>>>(lidx, ecnt, elist);
  k_estage<<<4096, 128, 0, stream>>>(x, kern0, k2x2, k3_w, k4_w, k5_w, k5_b,
                                     lidx, ridx, pidx, kernW);
  k_agg<<<1024, 128, 0, stream>>>(x, kernW, ridx, ecnt, elist, gc_w, gc_b, yg);
  k_head<<<64, 128, 0, stream>>>(yg, r0_w, hpart);
  k_hred<<<64, 256, 0, stream>>>(hpart, x2, r1_w, r1_b, capr2, h);
  k_wt<<<64, 256, 0, stream>>>(h, r3_w, r3_b, wt);
  k_final<<<64, 128, 0, stream>>>(wt, yg, x2, l1_w, l1_b, l3_w, l3_b, out);
}